// s_gcn_32641751449771
// MI455X (gfx1250) — compile-verified
//
#include <hip/hip_runtime.h>

typedef __attribute__((ext_vector_type(2))) float v2f;
typedef __attribute__((ext_vector_type(8))) float v8f;

#define KNUM 3
#define CIN  64
#define COUT 64
#define TDIM 2048
#define V    25
#define VP   32      // v padded to 32
#define TT   2       // t-values per block
#define NCOLS (TT*VP)

__global__ __launch_bounds__(256) void sgcn_wmma_kernel(
    const float* __restrict__ x,    // [16][64][2048][25]
    const float* __restrict__ A,    // [3][25][25]
    const float* __restrict__ W,    // [192][64]
    const float* __restrict__ b,    // [192]
    float* __restrict__ out)        // [16][64][2048][25]
{
    __shared__ __align__(16) float As[KNUM][VP][VP];       // 12 KB, zero padded
    __shared__ __align__(16) float Hs[KNUM*COUT][NCOLS];   // 48 KB

    const int tid  = threadIdx.x;
    const int lane = tid & 31;          // wave32
    const int l16  = lane & 15;
    const int half = lane >> 4;         // 0: lanes 0-15, 1: lanes 16-31
    const int wid  = tid >> 5;          // wave 0..7
    const int q    = wid & 1;           // which local t (column half of H)
    const int r    = wid >> 1;          // c-tile 0..3

    const int bid = blockIdx.x;
    const int n   = bid >> 10;          // 2048/TT = 1024 chunks per n
    const int t0  = (bid & 1023) * TT;
    const int t   = t0 + q;             // this wave's t

    // ---- stage zero-padded adjacency into LDS ----
    for (int i = tid; i < KNUM * VP * VP; i += 256) {
        const int k  = i / (VP * VP);
        const int vv = (i / VP) & (VP - 1);
        const int ww = i & (VP - 1);
        As[k][vv][ww] = (vv < V && ww < V) ? A[(k * V + vv) * V + ww] : 0.0f;
    }
    __syncthreads();

    const v8f vzero = {0.f, 0.f, 0.f, 0.f, 0.f, 0.f, 0.f, 0.f};

    // =========== GEMM1: H[k*64+c][v] = sum_c' W[k*64+c][c'] * x[n][c'][t][v] ===========
    v8f acc1[KNUM][2];
    #pragma unroll
    for (int k = 0; k < KNUM; ++k) { acc1[k][0] = vzero; acc1[k][1] = vzero; }

    const int xnt = n * (CIN * TDIM * V) + t * V;   // + c*(TDIM*V) + v

    // j=1 column index: v = 16 + l16; pad lanes (v >= 25) load a clamped
    // in-bounds address unconditionally and are zeroed by multiply-mask.
    // This keeps the load stream branch-free (no EXEC diamonds in the hot loop).
    const int   v1    = 16 + l16;
    const int   v1s   = (v1 < V) ? v1 : 0;
    const float mask1 = (v1 < V) ? 1.0f : 0.0f;

    #pragma unroll
    for (int kk = 0; kk < CIN / 4; ++kk) {
        const int c0    = kk * 4 + half * 2;        // K-slice per lane half
        const int base0 = xnt + c0 * (TDIM * V);
        const int base1 = base0 + (TDIM * V);
        // B operand (X): VGPR0 = X[c0][col], VGPR1 = X[c0+1][col]
        const v2f xf0 = (v2f){ x[base0 + l16],
                               x[base1 + l16] };            // v = l16 < 25 always
        const v2f xf1 = (v2f){ x[base0 + v1s] * mask1,
                               x[base1 + v1s] * mask1 };
        #pragma unroll
        for (int k = 0; k < KNUM; ++k) {
            const int row = k * COUT + r * 16 + l16;
            // A operand (W): lane holds 2 consecutive K values of its M row
            const v2f wf = *(const v2f*)(W + row * CIN + c0);
            acc1[k][0] = __builtin_amdgcn_wmma_f32_16x16x4_f32(
                false, wf, false, xf0, (short)0, acc1[k][0], false, false);
            acc1[k][1] = __builtin_amdgcn_wmma_f32_16x16x4_f32(
                false, wf, false, xf1, (short)0, acc1[k][1], false, false);
        }
    }

    // ---- add bias, stash H into LDS (C/D layout -> row major) ----
    #pragma unroll
    for (int k = 0; k < KNUM; ++k) {
        #pragma unroll
        for (int j = 0; j < 2; ++j) {
            #pragma unroll
            for (int i = 0; i < 8; ++i) {
                const int row = k * COUT + r * 16 + i + half * 8;
                const int col = q * VP + j * 16 + l16;
                Hs[row][col] = acc1[k][j][i] + b[row];
            }
        }
    }
    __syncthreads();

    // =========== GEMM2: out[c][w] = sum_k sum_v H[k*64+c][v] * A[k][v][w] ===========
    v8f acc2[2];
    acc2[0] = vzero; acc2[1] = vzero;

    #pragma unroll
    for (int k = 0; k < KNUM; ++k) {
        #pragma unroll
        for (int kk = 0; kk < VP / 4; ++kk) {
            const int v0 = kk * 4 + half * 2;
            // A operand (H): lane = M row (c), 2 consecutive K (v) values
            const v2f hf = *(const v2f*)(&Hs[k * COUT + r * 16 + l16][q * VP + v0]);
            #pragma unroll
            for (int j = 0; j < 2; ++j) {
                // B operand (padded adjacency): K rows striped over lane halves
                const v2f af = (v2f){ As[k][v0    ][j * 16 + l16],
                                      As[k][v0 + 1][j * 16 + l16] };
                acc2[j] = __builtin_amdgcn_wmma_f32_16x16x4_f32(
                    false, hf, false, af, (short)0, acc2[j], false, false);
            }
        }
    }

    // ---- store output tile: out[n][c][t][w], only w < 25 ----
    #pragma unroll
    for (int j = 0; j < 2; ++j) {
        const int w = j * 16 + l16;
        if (w < V) {
            #pragma unroll
            for (int i = 0; i < 8; ++i) {
                const int c = r * 16 + i + half * 8;
                out[((n * COUT + c) * TDIM + t) * V + w] = acc2[j][i];
            }
        }
    }
}

extern "C" void kernel_launch(void* const* d_in, const int* in_sizes, int n_in,
                              void* d_out, int out_size, void* d_ws, size_t ws_size,
                              hipStream_t stream) {
    const float* x = (const float*)d_in[0];
    const float* A = (const float*)d_in[1];
    const float* W = (const float*)d_in[2];
    const float* b = (const float*)d_in[3];
    float* out = (float*)d_out;

    const int blocks = 16 * (TDIM / TT);   // 16384 chunks of (n, 2 t-values)
    sgcn_wmma_kernel<<<dim3(blocks), dim3(256), 0, stream>>>(x, A, W, b, out);
}